// GAT_69458211111265
// MI455X (gfx1250) — compile-verified
//
#include <hip/hip_runtime.h>

#define NN 50000
#define EE 1600000
#define FF 128
#define HH 4
#define CC 32
#define GG 64

typedef float v2f __attribute__((ext_vector_type(2)));
typedef float v8f __attribute__((ext_vector_type(8)));

// ---- monotonic float<->uint order mapping for atomicMax on floats ----
__device__ __forceinline__ unsigned fenc(float f) {
    unsigned u = __float_as_uint(f);
    return (u & 0x80000000u) ? ~u : (u | 0x80000000u);
}
__device__ __forceinline__ float fdec(unsigned k) {
    unsigned u = (k & 0x80000000u) ? (k ^ 0x80000000u) : ~k;
    return __uint_as_float(u);
}
__device__ __forceinline__ float lrelu02(float x) { return x > 0.f ? x : 0.2f * x; }

__global__ __launch_bounds__(256) void fill_u32_kernel(unsigned* __restrict__ p, unsigned v, int n) {
    int i = blockIdx.x * 256 + threadIdx.x;
    if (i < n) p[i] = v;
}

// ---------------------------------------------------------------------
// D[N,128] = A[N,128] @ B[128,128] in exact fp32 via V_WMMA_F32_16X16X4_F32
// block = 256 threads = 8 waves; each wave owns one 16x16 output tile.
// A frag (16x4 f32, 2 VGPR): lanes 0-15 -> K=k,k+1 ; lanes 16-31 -> K=k+2,k+3
// B frag (4x16 f32, 2 VGPR): same K split, N = lane&15
// C/D (16x16 f32, 8 VGPR):  VGPR i -> row i (lanes 0-15) / row i+8 (lanes 16-31)
// ---------------------------------------------------------------------
__global__ __launch_bounds__(256) void gemm_node_kernel(const float* __restrict__ A,
                                                        const float* __restrict__ B,
                                                        float* __restrict__ D) {
    const int lane = threadIdx.x & 31;
    const int wave = threadIdx.x >> 5;      // 0..7 -> column tile
    const int row0 = blockIdx.x << 4;       // 16-row tile (N % 16 == 0)
    const int col0 = wave << 4;
    const int hi   = lane >> 4;             // 0: low K pair, 1: high K pair
    const int lr   = lane & 15;
    const float* arow = A + (size_t)(row0 + lr) * FF;
    const float* bcol = B + col0 + lr;
    v8f c = {};
#pragma unroll
    for (int k = 0; k < FF; k += 4) {
        const int ka = k + hi * 2;
        v2f a, b;
        a.x = arow[ka];
        a.y = arow[ka + 1];
        b.x = bcol[(size_t)ka * FF];
        b.y = bcol[(size_t)(ka + 1) * FF];
        c = __builtin_amdgcn_wmma_f32_16x16x4_f32(false, a, false, b, (short)0, c, false, false);
    }
    float* o = D + (size_t)row0 * FF + col0;
#pragma unroll
    for (int i = 0; i < 8; ++i)
        o[(size_t)(i + hi * 8) * FF + lr] = c[i];
}

// per-(node,head) attention coefficients: al_s = <h[n,h,:], a_src[h,:]>
__global__ __launch_bounds__(256) void alpha_kernel(const float* __restrict__ h,
                                                    const float* __restrict__ a_src,
                                                    const float* __restrict__ a_dst,
                                                    float* __restrict__ al_s,
                                                    float* __restrict__ al_d) {
    int t = blockIdx.x * 256 + threadIdx.x;
    if (t >= NN * HH) return;
    int n = t >> 2, head = t & 3;
    const float* hp = h + (size_t)n * FF + head * CC;
    const float* as = a_src + head * CC;
    const float* ad = a_dst + head * CC;
    float ss = 0.f, sd = 0.f;
#pragma unroll
    for (int c = 0; c < CC; ++c) { float v = hp[c]; ss += v * as[c]; sd += v * ad[c]; }
    al_s[t] = ss;
    al_d[t] = sd;
}

__device__ __forceinline__ void edge_sd(const int* __restrict__ src, const int* __restrict__ dst,
                                        int e, int& s, int& d) {
    if (e < EE) { s = src[e]; d = dst[e]; }
    else        { s = d = e - EE; }          // appended self-loops
}

// pass 1: segment max of leaky-relu logits over incoming edges per (dst,head)
__global__ __launch_bounds__(256) void edge_max_kernel(const int* __restrict__ src, const int* __restrict__ dst,
                                                       const float* __restrict__ al_s, const float* __restrict__ al_d,
                                                       unsigned* __restrict__ mkey) {
    int t = blockIdx.x * 256 + threadIdx.x;
    if (t >= (EE + NN) * HH) return;
    int e = t >> 2, head = t & 3, s, d;
    edge_sd(src, dst, e, s, d);
    float lg = lrelu02(al_s[s * HH + head] + al_d[d * HH + head]);
    atomicMax(&mkey[d * HH + head], fenc(lg));
}

// pass 2: segment sum of exp(logit - max)
__global__ __launch_bounds__(256) void edge_expsum_kernel(const int* __restrict__ src, const int* __restrict__ dst,
                                                          const float* __restrict__ al_s, const float* __restrict__ al_d,
                                                          const unsigned* __restrict__ mkey, float* __restrict__ ssum) {
    int t = blockIdx.x * 256 + threadIdx.x;
    if (t >= (EE + NN) * HH) return;
    int e = t >> 2, head = t & 3, s, d;
    edge_sd(src, dst, e, s, d);
    int idx = d * HH + head;
    float lg = lrelu02(al_s[s * HH + head] + al_d[idx]);
    atomicAdd(&ssum[idx], __expf(lg - fdec(mkey[idx])));
}

// pass 3: agg[dst] += alpha * h[src] ; one wave per edge, 4 channels/lane
__global__ __launch_bounds__(256) void edge_agg_kernel(const int* __restrict__ src, const int* __restrict__ dst,
                                                       const float* __restrict__ al_s, const float* __restrict__ al_d,
                                                       const unsigned* __restrict__ mkey, const float* __restrict__ ssum,
                                                       const float* __restrict__ h, float* __restrict__ agg) {
    int t = blockIdx.x * 256 + threadIdx.x;
    int e = t >> 5;
    if (e >= EE + NN) return;
    int lane = t & 31, s, d;
    edge_sd(src, dst, e, s, d);
    int head = lane >> 3;                      // 8 lanes (32 ch) per head
    int idx = d * HH + head;
    float lg = lrelu02(al_s[s * HH + head] + al_d[idx]);
    float alpha = __expf(lg - fdec(mkey[idx])) / ssum[idx];
    const float4 hv = *(const float4*)(h + (size_t)s * FF + lane * 4);
    float* o = agg + (size_t)d * FF + lane * 4;
    atomicAdd(o + 0, alpha * hv.x);
    atomicAdd(o + 1, alpha * hv.y);
    atomicAdd(o + 2, alpha * hv.z);
    atomicAdd(o + 3, alpha * hv.w);
}

// h = relu(agg + b); per-(graph,column) running max
__global__ __launch_bounds__(256) void bias_relu_colmax_kernel(float* __restrict__ agg,
                                                               const float* __restrict__ bias,
                                                               const int* __restrict__ batch,
                                                               unsigned* __restrict__ cmax) {
    int t = blockIdx.x * 256 + threadIdx.x;
    if (t >= NN * FF) return;
    int n = t >> 7, j = t & 127;
    float v = fmaxf(agg[t] + bias[j], 0.f);
    agg[t] = v;
    atomicMax(&cmax[batch[n] * FF + j], fenc(v));
}

// e = exp(h - colmax); stash e; colsum += e
__global__ __launch_bounds__(256) void col_expsum_kernel(float* __restrict__ agg,
                                                         const int* __restrict__ batch,
                                                         const unsigned* __restrict__ cmax,
                                                         float* __restrict__ csum) {
    int t = blockIdx.x * 256 + threadIdx.x;
    if (t >= NN * FF) return;
    int n = t >> 7, j = t & 127;
    int gi = batch[n] * FF + j;
    float e = __expf(agg[t] - fdec(cmax[gi]));
    agg[t] = e;
    atomicAdd(&csum[gi], e);
}

// h_sm = e / colsum (in place); gate logit per node; per-graph gate max
__global__ __launch_bounds__(256) void gate_kernel(float* __restrict__ agg, const int* __restrict__ batch,
                                                   const float* __restrict__ csum,
                                                   const float* __restrict__ gate_w, const float* __restrict__ gate_b,
                                                   float* __restrict__ gate, unsigned* __restrict__ gmax) {
    int t = blockIdx.x * 256 + threadIdx.x;
    int n = t >> 5;
    if (n >= NN) return;
    int lane = t & 31;
    int g = batch[n];
    float4 e  = *(const float4*)(agg  + (size_t)n * FF + lane * 4);
    float4 cs = *(const float4*)(csum + (size_t)g * FF + lane * 4);
    float4 hs; hs.x = e.x / cs.x; hs.y = e.y / cs.y; hs.z = e.z / cs.z; hs.w = e.w / cs.w;
    *(float4*)(agg + (size_t)n * FF + lane * 4) = hs;
    float4 gw = *(const float4*)(gate_w + lane * 4);
    float p = hs.x * gw.x + hs.y * gw.y + hs.z * gw.z + hs.w * gw.w;
#pragma unroll
    for (int m = 16; m >= 1; m >>= 1) p += __shfl_xor(p, m, 32);
    if (lane == 0) {
        float gv = p + gate_b[0];
        gate[n] = gv;
        atomicMax(&gmax[g], fenc(gv));
    }
}

__global__ __launch_bounds__(256) void gate_expsum_kernel(float* __restrict__ gate, const int* __restrict__ batch,
                                                          const unsigned* __restrict__ gmax, float* __restrict__ gsum) {
    int n = blockIdx.x * 256 + threadIdx.x;
    if (n >= NN) return;
    int g = batch[n];
    float ge = __expf(gate[n] - fdec(gmax[g]));
    gate[n] = ge;
    atomicAdd(&gsum[g], ge);
}

// hg[g,:] += (gate_sm[n]) * h_sm[n,:]
__global__ __launch_bounds__(256) void pool_kernel(const float* __restrict__ agg, const float* __restrict__ gate,
                                                   const int* __restrict__ batch, const float* __restrict__ gsum,
                                                   float* __restrict__ hg) {
    int t = blockIdx.x * 256 + threadIdx.x;
    if (t >= NN * FF) return;
    int n = t >> 7, j = t & 127;
    int g = batch[n];
    float w = gate[n] / gsum[g];
    atomicAdd(&hg[g * FF + j], w * agg[t]);
}

// out = relu(hg @ lin_w + lin_b) @ cls_w + cls_b      [64,128]->[64,64]->[64,10]
__global__ __launch_bounds__(256) void head_kernel(const float* __restrict__ hg,
                                                   const float* __restrict__ lin_w, const float* __restrict__ lin_b,
                                                   const float* __restrict__ cls_w, const float* __restrict__ cls_b,
                                                   float* __restrict__ out) {
    __shared__ float tmp[GG * 64];
    int tid = threadIdx.x;
    for (int idx = tid; idx < GG * 64; idx += 256) {
        int g = idx >> 6, j = idx & 63;
        float acc = lin_b[j];
#pragma unroll 8
        for (int k = 0; k < FF; ++k) acc += hg[g * FF + k] * lin_w[k * 64 + j];
        tmp[idx] = fmaxf(acc, 0.f);
    }
    __syncthreads();
    for (int idx = tid; idx < GG * 10; idx += 256) {
        int g = idx / 10, o = idx % 10;
        float acc = cls_b[o];
#pragma unroll 8
        for (int k = 0; k < 64; ++k) acc += tmp[g * 64 + k] * cls_w[k * 10 + o];
        out[idx] = acc;
    }
}

extern "C" void kernel_launch(void* const* d_in, const int* in_sizes, int n_in,
                              void* d_out, int out_size, void* d_ws, size_t ws_size,
                              hipStream_t stream) {
    const float* x      = (const float*)d_in[0];
    const int*   ei     = (const int*)d_in[1];
    const int*   src    = ei;
    const int*   dst    = ei + EE;
    const int*   batch  = (const int*)d_in[2];
    const float* W[2]    = {(const float*)d_in[3], (const float*)d_in[7]};
    const float* Asrc[2] = {(const float*)d_in[4], (const float*)d_in[8]};
    const float* Adst[2] = {(const float*)d_in[5], (const float*)d_in[9]};
    const float* Bias[2] = {(const float*)d_in[6], (const float*)d_in[10]};
    const float* gate_w = (const float*)d_in[11];
    const float* gate_b = (const float*)d_in[12];
    const float* lin_w  = (const float*)d_in[13];
    const float* lin_b  = (const float*)d_in[14];
    const float* cls_w  = (const float*)d_in[15];
    const float* cls_b  = (const float*)d_in[16];
    float* out = (float*)d_out;

    // ---- workspace carve-out (256B aligned) ----
    char* wsb = (char*)d_ws;
    size_t off = 0;
    auto take = [&](size_t elems) {
        void* p = (void*)(wsb + off);
        off += ((elems * 4 + 255) & ~(size_t)255);
        return p;
    };
    float*    hbuf = (float*)   take((size_t)NN * FF);   // h = in @ W
    float*    agg  = (float*)   take((size_t)NN * FF);   // aggregate -> relu -> exp -> h_sm
    float*    al_s = (float*)   take((size_t)NN * HH);
    float*    al_d = (float*)   take((size_t)NN * HH);
    unsigned* mkey = (unsigned*)take((size_t)NN * HH);
    float*    ssum = (float*)   take((size_t)NN * HH);
    unsigned* cmax = (unsigned*)take((size_t)GG * FF);
    float*    csum = (float*)   take((size_t)GG * FF);
    float*    gate = (float*)   take((size_t)NN);
    unsigned* gmax = (unsigned*)take((size_t)GG);
    float*    gsum = (float*)   take((size_t)GG);
    float*    hg   = (float*)   take((size_t)GG * FF);

    const unsigned KNEG_INF = 0x007FFFFFu;   // fenc(-inf)
    auto fill = [&](void* p, unsigned v, int n) {
        fill_u32_kernel<<<(n + 255) / 256, 256, 0, stream>>>((unsigned*)p, v, n);
    };

    fill(hg, 0u, GG * FF);

    const float* in = x;
    const int ENH = (EE + NN) * HH;
    const int EW  = (EE + NN) * 32;
    for (int L = 0; L < 2; ++L) {
        gemm_node_kernel<<<NN / 16, 256, 0, stream>>>(in, W[L], hbuf);
        alpha_kernel<<<(NN * HH + 255) / 256, 256, 0, stream>>>(hbuf, Asrc[L], Adst[L], al_s, al_d);
        // agg re-zero happens only after the GEMM consumed it (stream ordered)
        fill(agg, 0u, NN * FF);
        fill(mkey, KNEG_INF, NN * HH);
        fill(ssum, 0u, NN * HH);
        edge_max_kernel   <<<(ENH + 255) / 256, 256, 0, stream>>>(src, dst, al_s, al_d, mkey);
        edge_expsum_kernel<<<(ENH + 255) / 256, 256, 0, stream>>>(src, dst, al_s, al_d, mkey, ssum);
        edge_agg_kernel   <<<(EW  + 255) / 256, 256, 0, stream>>>(src, dst, al_s, al_d, mkey, ssum, hbuf, agg);
        fill(cmax, KNEG_INF, GG * FF);
        fill(csum, 0u, GG * FF);
        bias_relu_colmax_kernel<<<(NN * FF) / 256, 256, 0, stream>>>(agg, Bias[L], batch, cmax);
        col_expsum_kernel      <<<(NN * FF) / 256, 256, 0, stream>>>(agg, batch, cmax, csum);
        fill(gmax, KNEG_INF, GG);
        fill(gsum, 0u, GG);
        gate_kernel       <<<(NN * 32) / 256, 256, 0, stream>>>(agg, batch, csum, gate_w, gate_b, gate, gmax);
        gate_expsum_kernel<<<(NN + 255) / 256, 256, 0, stream>>>(gate, batch, gmax, gsum);
        pool_kernel       <<<(NN * FF) / 256, 256, 0, stream>>>(agg, gate, batch, gsum, hg);
        in = agg;   // layer-2 input = seg-softmaxed features of layer 1
    }
    head_kernel<<<1, 256, 0, stream>>>(hg, lin_w, lin_b, cls_w, cls_b, out);
}